// SparseAxialCausalAttention_32255204393235
// MI455X (gfx1250) — compile-verified
//
#include <hip/hip_runtime.h>

typedef float v2f __attribute__((ext_vector_type(2)));
typedef float v8f __attribute__((ext_vector_type(8)));

#define NTOK    4351
#define SEQ_N   4352
#define DIMX    512
#define QKVC    1536
#define HH      8
#define DHH     64
#define TEXTL   256
#define IMGW    64
#define BB      4
#define BHH     32
#define NROWS   17408
#define NEGMAX  (-3.402823466e38f)

// ---- CDNA5 async global->LDS staging (guarded; falls back to ds_store) ----
#if defined(__has_builtin)
#if __has_builtin(__builtin_amdgcn_global_load_async_to_lds_b128)
#define HAVE_ASYNC_LDS 1
#endif
#endif

#if defined(HAVE_ASYNC_LDS)
// Builtin prototype (from clang diagnostic): param0 = int4* addrspace(1),
// param1 = int4* addrspace(3), then imm offset, imm cpol.
typedef int v4i_vs __attribute__((vector_size(16)));
typedef __attribute__((address_space(1))) v4i_vs GV4;
typedef __attribute__((address_space(3))) v4i_vs LV4;
#if __has_builtin(__builtin_amdgcn_s_wait_asynccnt)
#define WAIT_ASYNC(n) __builtin_amdgcn_s_wait_asynccnt(n)
#else
#define WAIT_ASYNC(n) asm volatile("s_wait_asynccnt " #n ::: "memory")
#endif
#else
#define WAIT_ASYNC(n) do {} while (0)
#endif

// Copy 16 contiguous floats (64B) global -> LDS for this thread.
__device__ __forceinline__ void stage16(const float* __restrict__ gsrc, float* ldst) {
#if defined(HAVE_ASYNC_LDS)
  __builtin_amdgcn_global_load_async_to_lds_b128((GV4*)(gsrc + 0),  (LV4*)(ldst + 0),  0, 0);
  __builtin_amdgcn_global_load_async_to_lds_b128((GV4*)(gsrc + 4),  (LV4*)(ldst + 4),  0, 0);
  __builtin_amdgcn_global_load_async_to_lds_b128((GV4*)(gsrc + 8),  (LV4*)(ldst + 8),  0, 0);
  __builtin_amdgcn_global_load_async_to_lds_b128((GV4*)(gsrc + 12), (LV4*)(ldst + 12), 0, 0);
#else
  const float4* gs = (const float4*)gsrc;
  float4* ld = (float4*)ldst;
  ld[0] = gs[0]; ld[1] = gs[1]; ld[2] = gs[2]; ld[3] = gs[3];
#endif
}

// D = A(16x4) * B(4x16) + C, fp32 WMMA.
__device__ __forceinline__ v8f wmma4(v2f a, v2f b, v8f c) {
  return __builtin_amdgcn_wmma_f32_16x16x4_f32(false, a, false, b, (short)0, c, false, false);
}

// Row-wise reductions across the 16 lanes of each half-wave (C-layout rows).
__device__ __forceinline__ float rowmax16(float x) {
  x = fmaxf(x, __shfl_xor(x, 1, 32));
  x = fmaxf(x, __shfl_xor(x, 2, 32));
  x = fmaxf(x, __shfl_xor(x, 4, 32));
  x = fmaxf(x, __shfl_xor(x, 8, 32));
  return x;
}
__device__ __forceinline__ float rowsum16(float x) {
  x += __shfl_xor(x, 1, 32);
  x += __shfl_xor(x, 2, 32);
  x += __shfl_xor(x, 4, 32);
  x += __shfl_xor(x, 8, 32);
  return x;
}

// ---------------------------------------------------------------------------
// Kernel 1: qkv = pad(x) @ w_qkv, head-split scatter. B tile double-buffered
// in LDS via async copies; 4 waves x (32 rows x 64 cols) per block.
// ---------------------------------------------------------------------------
__global__ __launch_bounds__(128) void qkv_proj_kernel(
    const float* __restrict__ x, const float* __restrict__ w,
    float* __restrict__ qws, float* __restrict__ kws, float* __restrict__ vws)
{
  __shared__ float bbuf[2][32 * 64];   // 16 KB double buffer (32 K-rows x 64 cols)
  const int tid  = threadIdx.x;
  const int lane = tid & 31;
  const int wid  = tid >> 5;
  const int lm = lane & 15;
  const int hi = lane >> 4;
  const int kp = hi << 1;
  const int tn = blockIdx.x % (QKVC / 64);
  const int tm = blockIdx.x / (QKVC / 64);
  const int rbase = tm * 128 + wid * 32;
  const int cbase = tn * 64;

  // staging assignment: thread copies 16 floats of each 32x64 chunk
  const int srow = tid >> 2;              // 0..31
  const int scol = (tid & 3) * 16;        // 0/16/32/48
  const float* wsrc = w + (size_t)srow * QKVC + cbase + scol;
  float* ld0 = &bbuf[0][srow * 64 + scol];
  float* ld1 = &bbuf[1][srow * 64 + scol];

  v8f acc[2][4];
  #pragma unroll
  for (int i = 0; i < 2; ++i)
    #pragma unroll
    for (int jt = 0; jt < 4; ++jt)
      #pragma unroll
      for (int j = 0; j < 8; ++j) acc[i][jt][j] = 0.0f;

  const int r0 = rbase + lm;
  const int r1 = r0 + 16;
  const int b0 = r0 / SEQ_N, t0 = r0 % SEQ_N;
  const int b1 = r1 / SEQ_N, t1 = r1 % SEQ_N;
  const float f0 = (t0 < NTOK) ? 1.0f : 0.0f;   // zero the pad row
  const float f1 = (t1 < NTOK) ? 1.0f : 0.0f;
  const float* xr0 = x + (size_t)(b0 * NTOK + (t0 < NTOK ? t0 : 0)) * DIMX;
  const float* xr1 = x + (size_t)(b1 * NTOK + (t1 < NTOK ? t1 : 0)) * DIMX;

  stage16(wsrc, ld0);   // prime chunk 0

  for (int c = 0; c < 16; ++c) {
    const float* bl = bbuf[c & 1];
    if (c + 1 < 16) {
      stage16(wsrc + (size_t)(c + 1) * 32 * QKVC, (c & 1) ? ld0 : ld1);
      WAIT_ASYNC(4);          // chunk c's 4 ops done (async loads retire in order)
    } else {
      WAIT_ASYNC(0);
    }
    __syncthreads();
    const int kbase = c * 32;
    #pragma unroll
    for (int k2 = 0; k2 < 32; k2 += 4) {
      const int krel = k2 + kp;
      const int kg = kbase + krel;
      v2f a0, a1;
      a0.x = xr0[kg] * f0;  a0.y = xr0[kg + 1] * f0;
      a1.x = xr1[kg] * f1;  a1.y = xr1[kg + 1] * f1;
      const float* brow0 = bl + krel * 64 + lm;
      const float* brow1 = brow0 + 64;
      #pragma unroll
      for (int jt = 0; jt < 4; ++jt) {
        v2f bv; bv.x = brow0[jt * 16]; bv.y = brow1[jt * 16];
        acc[0][jt] = wmma4(a0, bv, acc[0][jt]);
        acc[1][jt] = wmma4(a1, bv, acc[1][jt]);
      }
    }
    __syncthreads();   // buffer may be overwritten next iteration
  }

  #pragma unroll
  for (int i = 0; i < 2; ++i) {
    const int rr = rbase + i * 16 + hi * 8;   // SEQ_N % 8 == 0 -> no batch crossing
    const int bidx = rr / SEQ_N;
    const int t = rr % SEQ_N;
    #pragma unroll
    for (int jt = 0; jt < 4; ++jt) {
      const int c = cbase + jt * 16 + lm;
      const int which = c >> 9;          // 0=q 1=k 2=v
      const int h = (c >> 6) & 7;
      const int d = c & 63;
      float* dst = (which == 0) ? qws : ((which == 1) ? kws : vws);
      const float sc = (which == 0) ? 0.125f : 1.0f;   // DH^-0.5
      const size_t base = ((size_t)(bidx * HH + h) * SEQ_N + t) * DHH + d;
      #pragma unroll
      for (int j = 0; j < 8; ++j)
        dst[base + (size_t)j * DHH] = acc[i][jt][j] * sc;
    }
  }
}

// ---------------------------------------------------------------------------
// Kernel 2: causal text attention, 32 heads x 256 queries. Flash-style.
// 4 waves/block, each wave owns 32 query rows. P tile round-trips LDS.
// ---------------------------------------------------------------------------
__global__ __launch_bounds__(128) void text_attn_kernel(
    const float* __restrict__ q, const float* __restrict__ k,
    const float* __restrict__ v, float* __restrict__ o)
{
  __shared__ float plds[4][512];   // per-wave private 32x16 P tile
  const int lane = threadIdx.x & 31;
  const int wid  = threadIdx.x >> 5;
  const int lm = lane & 15;
  const int hi = lane >> 4;
  const int kp = hi << 1;
  const int head = blockIdx.x >> 1;
  const int qbase = (blockIdx.x & 1) * 128 + wid * 32;
  const size_t hb = (size_t)head * SEQ_N * DHH;
  const float* Q = q + hb;
  const float* K = k + hb;
  const float* V = v + hb;
  const float* qr0 = Q + (size_t)(qbase + lm) * DHH;
  const float* qr1 = qr0 + 16 * DHH;

  v8f oacc[2][4];
  float rmax[2][8], rsum[2][8];
  #pragma unroll
  for (int i = 0; i < 2; ++i)
    #pragma unroll
    for (int j = 0; j < 8; ++j) {
      rmax[i][j] = NEGMAX; rsum[i][j] = 0.0f;
      #pragma unroll
      for (int ct = 0; ct < 4; ++ct) oacc[i][ct][j] = 0.0f;
    }

  const int lasttile = (qbase + 31) >> 4;   // wave-uniform causal tile bound
  for (int jt = 0; jt <= lasttile; ++jt) {
    const int k0 = jt * 16;
    v8f s[2];
    #pragma unroll
    for (int i = 0; i < 2; ++i)
      #pragma unroll
      for (int j = 0; j < 8; ++j) s[i][j] = 0.0f;
    const float* kr = K + (size_t)(k0 + lm) * DHH;
    #pragma unroll
    for (int s4 = 0; s4 < 16; ++s4) {
      v2f bv; bv.x = kr[4 * s4 + kp];  bv.y = kr[4 * s4 + kp + 1];
      v2f a0; a0.x = qr0[4 * s4 + kp]; a0.y = qr0[4 * s4 + kp + 1];
      v2f a1; a1.x = qr1[4 * s4 + kp]; a1.y = qr1[4 * s4 + kp + 1];
      s[0] = wmma4(a0, bv, s[0]);
      s[1] = wmma4(a1, bv, s[1]);
    }
    const int kn = k0 + lm;   // this lane's key column
    #pragma unroll
    for (int i = 0; i < 2; ++i) {
      #pragma unroll
      for (int j = 0; j < 8; ++j) {
        const int qrow = qbase + i * 16 + hi * 8 + j;
        const float val = (kn <= qrow) ? s[i][j] : NEGMAX;
        const float rm = rowmax16(val);
        const float newm = fmaxf(rmax[i][j], rm);
        const float corr = __expf(rmax[i][j] - newm);
        rmax[i][j] = newm;
        const float p = __expf(val - newm);
        rsum[i][j] = rsum[i][j] * corr + rowsum16(p);
        #pragma unroll
        for (int ct = 0; ct < 4; ++ct) oacc[i][ct][j] *= corr;
        plds[wid][(i * 16 + hi * 8 + j) * 16 + lm] = p;
      }
    }
    asm volatile("s_wait_dscnt 0x0" ::: "memory");   // cross-lane LDS store->load
    #pragma unroll
    for (int s4 = 0; s4 < 4; ++s4) {
      v2f a0, a1;
      a0.x = plds[wid][lm * 16 + 4 * s4 + kp];
      a0.y = plds[wid][lm * 16 + 4 * s4 + kp + 1];
      a1.x = plds[wid][(16 + lm) * 16 + 4 * s4 + kp];
      a1.y = plds[wid][(16 + lm) * 16 + 4 * s4 + kp + 1];
      const float* vr = V + (size_t)(k0 + 4 * s4 + kp) * DHH + lm;
      #pragma unroll
      for (int ct = 0; ct < 4; ++ct) {
        v2f bv;
        bv.x = vr[ct * 16];
        bv.y = vr[ct * 16 + DHH];
        oacc[0][ct] = wmma4(a0, bv, oacc[0][ct]);
        oacc[1][ct] = wmma4(a1, bv, oacc[1][ct]);
      }
    }
  }

  #pragma unroll
  for (int i = 0; i < 2; ++i)
    #pragma unroll
    for (int j = 0; j < 8; ++j) {
      const float inv = 1.0f / rsum[i][j];
      const int t = qbase + i * 16 + hi * 8 + j;
      #pragma unroll
      for (int ct = 0; ct < 4; ++ct)
        o[hb + (size_t)t * DHH + ct * 16 + lm] = oacc[i][ct][j] * inv;
    }
}

// ---------------------------------------------------------------------------
// Kernel 3: axial image attention. One block per (b,h,img_row); keys =
// 256 masked text keys + 64 causal image-row keys. 2 waves x 32 queries.
// ---------------------------------------------------------------------------
__global__ __launch_bounds__(64) void img_attn_kernel(
    const float* __restrict__ q, const float* __restrict__ k,
    const float* __restrict__ v, const unsigned char* __restrict__ mask,
    float* __restrict__ o)
{
  __shared__ float plds[2][512];
  const int lane = threadIdx.x & 31;
  const int wid  = threadIdx.x >> 5;
  const int lm = lane & 15;
  const int hi = lane >> 4;
  const int kp = hi << 1;
  const int bh = blockIdx.x >> 6;
  const int row = blockIdx.x & 63;
  const int b = bh >> 3;
  const size_t hb = (size_t)bh * SEQ_N * DHH;
  const float* Q = q + hb;
  const float* K = k + hb;
  const float* V = v + hb;
  const int imgbase = TEXTL + row * IMGW;
  const int qi0 = wid * 32;
  const float* qr0 = Q + (size_t)(imgbase + qi0 + lm) * DHH;
  const float* qr1 = qr0 + 16 * DHH;

  v8f oacc[2][4];
  float rmax[2][8], rsum[2][8];
  #pragma unroll
  for (int i = 0; i < 2; ++i)
    #pragma unroll
    for (int j = 0; j < 8; ++j) {
      rmax[i][j] = NEGMAX; rsum[i][j] = 0.0f;
      #pragma unroll
      for (int ct = 0; ct < 4; ++ct) oacc[i][ct][j] = 0.0f;
    }

  const int lasttile = 16 + ((qi0 + 31) >> 4);   // wave-uniform
  for (int jt = 0; jt <= lasttile; ++jt) {
    const bool istext = (jt < 16);
    const int jj = (jt - 16) * 16 + lm;           // img-relative key (if !istext)
    const int tilepos = istext ? (jt * 16) : (imgbase + (jt - 16) * 16);
    const float* kr = K + (size_t)(tilepos + lm) * DHH;
    v8f s[2];
    #pragma unroll
    for (int i = 0; i < 2; ++i)
      #pragma unroll
      for (int j = 0; j < 8; ++j) s[i][j] = 0.0f;
    #pragma unroll
    for (int s4 = 0; s4 < 16; ++s4) {
      v2f bv; bv.x = kr[4 * s4 + kp];  bv.y = kr[4 * s4 + kp + 1];
      v2f a0; a0.x = qr0[4 * s4 + kp]; a0.y = qr0[4 * s4 + kp + 1];
      v2f a1; a1.x = qr1[4 * s4 + kp]; a1.y = qr1[4 * s4 + kp + 1];
      s[0] = wmma4(a0, bv, s[0]);
      s[1] = wmma4(a1, bv, s[1]);
    }
    const bool colok = istext ? (mask[b * TEXTL + jt * 16 + lm] != 0) : true;
    #pragma unroll
    for (int i = 0; i < 2; ++i) {
      #pragma unroll
      for (int j = 0; j < 8; ++j) {
        const int qi = qi0 + i * 16 + hi * 8 + j;
        const bool ok = istext ? colok : (jj <= qi);
        const float val = ok ? s[i][j] : NEGMAX;
        const float rm = rowmax16(val);
        const float newm = fmaxf(rmax[i][j], rm);
        const float corr = __expf(rmax[i][j] - newm);
        rmax[i][j] = newm;
        const float p = __expf(val - newm);
        rsum[i][j] = rsum[i][j] * corr + rowsum16(p);
        #pragma unroll
        for (int ct = 0; ct < 4; ++ct) oacc[i][ct][j] *= corr;
        plds[wid][(i * 16 + hi * 8 + j) * 16 + lm] = p;
      }
    }
    asm volatile("s_wait_dscnt 0x0" ::: "memory");
    #pragma unroll
    for (int s4 = 0; s4 < 4; ++s4) {
      v2f a0, a1;
      a0.x = plds[wid][lm * 16 + 4 * s4 + kp];
      a0.y = plds[wid][lm * 16 + 4 * s4 + kp + 1];
      a1.x = plds[wid][(16 + lm) * 16 + 4 * s4 + kp];
      a1.y = plds[wid][(16 + lm) * 16 + 4 * s4 + kp + 1];
      const float* vr = V + (size_t)(tilepos + 4 * s4 + kp) * DHH + lm;
      #pragma unroll
      for (int ct = 0; ct < 4; ++ct) {
        v2f bv;
        bv.x = vr[ct * 16];
        bv.y = vr[ct * 16 + DHH];
        oacc[0][ct] = wmma4(a0, bv, oacc[0][ct]);
        oacc[1][ct] = wmma4(a1, bv, oacc[1][ct]);
      }
    }
  }

  #pragma unroll
  for (int i = 0; i < 2; ++i)
    #pragma unroll
    for (int j = 0; j < 8; ++j) {
      const float inv = 1.0f / rsum[i][j];
      const int t = imgbase + qi0 + i * 16 + hi * 8 + j;
      #pragma unroll
      for (int ct = 0; ct < 4; ++ct)
        o[hb + (size_t)t * DHH + ct * 16 + lm] = oacc[i][ct][j] * inv;
    }
}

// ---------------------------------------------------------------------------
// Kernel 4: out = concat_heads @ w_out + b_out (pad row dropped), with the
// same async-LDS double-buffered B tile as kernel 1.
// ---------------------------------------------------------------------------
__global__ __launch_bounds__(128) void out_proj_kernel(
    const float* __restrict__ hws, const float* __restrict__ w,
    const float* __restrict__ bias, float* __restrict__ out)
{
  __shared__ float bbuf[2][32 * 64];
  const int tid  = threadIdx.x;
  const int lane = tid & 31;
  const int wid  = tid >> 5;
  const int lm = lane & 15;
  const int hi = lane >> 4;
  const int kp = hi << 1;
  const int tn = blockIdx.x % (DIMX / 64);
  const int tm = blockIdx.x / (DIMX / 64);
  const int rbase = tm * 128 + wid * 32;
  const int cbase = tn * 64;

  const int srow = tid >> 2;
  const int scol = (tid & 3) * 16;
  const float* wsrc = w + (size_t)srow * DIMX + cbase + scol;
  float* ld0 = &bbuf[0][srow * 64 + scol];
  float* ld1 = &bbuf[1][srow * 64 + scol];

  v8f acc[2][4];
  #pragma unroll
  for (int i = 0; i < 2; ++i)
    #pragma unroll
    for (int jt = 0; jt < 4; ++jt)
      #pragma unroll
      for (int j = 0; j < 8; ++j) acc[i][jt][j] = 0.0f;

  const int r0 = rbase + lm, r1 = r0 + 16;
  const int b0 = r0 / SEQ_N, t0 = r0 % SEQ_N;
  const int b1 = r1 / SEQ_N, t1 = r1 % SEQ_N;
  const float* h0 = hws + ((size_t)(b0 * HH) * SEQ_N + t0) * DHH;
  const float* h1 = hws + ((size_t)(b1 * HH) * SEQ_N + t1) * DHH;

  stage16(wsrc, ld0);

  for (int c = 0; c < 16; ++c) {
    const float* bl = bbuf[c & 1];
    if (c + 1 < 16) {
      stage16(wsrc + (size_t)(c + 1) * 32 * DIMX, (c & 1) ? ld0 : ld1);
      WAIT_ASYNC(4);
    } else {
      WAIT_ASYNC(0);
    }
    __syncthreads();
    const int kbase = c * 32;
    #pragma unroll
    for (int k2 = 0; k2 < 32; k2 += 4) {
      const int krel = k2 + kp;
      const int kg = kbase + krel;
      const size_t off = (size_t)(kg >> 6) * SEQ_N * DHH + (kg & 63);
      v2f a0, a1;
      a0.x = h0[off]; a0.y = h0[off + 1];
      a1.x = h1[off]; a1.y = h1[off + 1];
      const float* brow0 = bl + krel * 64 + lm;
      const float* brow1 = brow0 + 64;
      #pragma unroll
      for (int jt = 0; jt < 4; ++jt) {
        v2f bv; bv.x = brow0[jt * 16]; bv.y = brow1[jt * 16];
        acc[0][jt] = wmma4(a0, bv, acc[0][jt]);
        acc[1][jt] = wmma4(a1, bv, acc[1][jt]);
      }
    }
    __syncthreads();
  }

  #pragma unroll
  for (int i = 0; i < 2; ++i) {
    const int rr = rbase + i * 16 + hi * 8;
    const int bidx = rr / SEQ_N;
    const int tt = rr % SEQ_N;
    #pragma unroll
    for (int jt = 0; jt < 4; ++jt) {
      const int c = cbase + jt * 16 + lm;
      const float bb = bias[c];
      #pragma unroll
      for (int j = 0; j < 8; ++j) {
        const int t = tt + j;
        if (t < NTOK)
          out[((size_t)bidx * NTOK + t) * DIMX + c] = acc[i][jt][j] + bb;
      }
    }
  }
}

extern "C" void kernel_launch(void* const* d_in, const int* in_sizes, int n_in,
                              void* d_out, int out_size, void* d_ws, size_t ws_size,
                              hipStream_t stream) {
  (void)in_sizes; (void)n_in; (void)out_size; (void)ws_size;
  const float* x            = (const float*)d_in[0];
  const unsigned char* mask = (const unsigned char*)d_in[1];  // bool (B,256)
  const float* w_qkv        = (const float*)d_in[2];
  const float* w_out        = (const float*)d_in[3];
  const float* b_out        = (const float*)d_in[4];
  float* out = (float*)d_out;

  const size_t per = (size_t)BHH * SEQ_N * DHH;   // 8,912,896 floats each
  float* qws = (float*)d_ws;
  float* kws = qws + per;
  float* vws = kws + per;
  float* ows = vws + per;

  qkv_proj_kernel<<<dim3((NROWS / 128) * (QKVC / 64)), dim3(128), 0, stream>>>(
      x, w_qkv, qws, kws, vws);
  text_attn_kernel<<<dim3(BHH * 2), dim3(128), 0, stream>>>(qws, kws, vws, ows);
  img_attn_kernel<<<dim3(BB * HH * IMGW), dim3(64), 0, stream>>>(
      qws, kws, vws, mask, ows);
  out_proj_kernel<<<dim3((NROWS / 128) * (DIMX / 64)), dim3(128), 0, stream>>>(
      ows, w_out, b_out, out);
}